// CombinedNodeFeaturesUwu_7919919694210
// MI455X (gfx1250) — compile-verified
//
#include <hip/hip_runtime.h>

typedef float v2f __attribute__((ext_vector_type(2)));
typedef float v8f __attribute__((ext_vector_type(8)));

// ---------------- utility kernels ----------------

__global__ void zero_f32_kernel(float* __restrict__ p, int n) {
    int i = blockIdx.x * blockDim.x + threadIdx.x;
    if (i < n) p[i] = 0.0f;
}

// Convert int64 edge indices -> int32, accumulate weighted in-degree over dst.
__global__ void prep_edges_kernel(const long long* __restrict__ ei,
                                  const float* __restrict__ w,
                                  int* __restrict__ src, int* __restrict__ dst,
                                  float* __restrict__ deg, int n_edges) {
    int e = blockIdx.x * blockDim.x + threadIdx.x;
    if (e >= n_edges) return;
    int s = (int)ei[e];
    int d = (int)ei[n_edges + e];
    src[e] = s;
    dst[e] = d;
    atomicAdd(&deg[d], w[e]);
}

__global__ void dinv_kernel(float* __restrict__ deg, int n) {
    int i = blockIdx.x * blockDim.x + threadIdx.x;
    if (i >= n) return;
    float d = deg[i];
    deg[i] = (d > 0.0f) ? rsqrtf(d) : 0.0f;   // deg buffer now holds dinv
}

__global__ void norm_kernel(const float* __restrict__ dinv,
                            const float* __restrict__ w,
                            const int* __restrict__ src, const int* __restrict__ dst,
                            float* __restrict__ norm, int n_edges) {
    int e = blockIdx.x * blockDim.x + threadIdx.x;
    if (e >= n_edges) return;
    norm[e] = dinv[src[e]] * w[e] * dinv[dst[e]];
}

// ---------------- WMMA GEMM: h = x @ W ----------------
// x: [n_rows, FI] row-major, W: [FI, FO] row-major, h: [n_rows, FO].
// One wave32 per 16x16 output tile using V_WMMA_F32_16X16X4_F32.
// Compile-time FI/FO: fully unrolled, no bounds checks, no exec divergence.
// n_rows % 16 == 0 and FO % 16 == 0 guaranteed by the problem shapes.
template <int FI, int FO>
__global__ void gemm_wmma_f32_kernel(const float* __restrict__ x,
                                     const float* __restrict__ W,
                                     float* __restrict__ h,
                                     int n_rows) {
    const int lane = threadIdx.x & 31;
    const int wave = blockIdx.x * (blockDim.x >> 5) + (threadIdx.x >> 5); // wave-uniform
    constexpr int NTN = FO / 16;
    const int total = (n_rows >> 4) * NTN;
    if (wave >= total) return;                 // wave-uniform: EXEC stays all-1s for WMMA

    const int mt   = wave / NTN;
    const int nt   = wave - mt * NTN;
    const int l15  = lane & 15;
    const int half = lane >> 4;                // 0: lanes 0-15, 1: lanes 16-31
    const int row  = mt * 16 + l15;            // A row for this lane
    const int col  = nt * 16 + l15;            // B/D column for this lane

    v8f c = {0.f, 0.f, 0.f, 0.f, 0.f, 0.f, 0.f, 0.f};

    if constexpr (FI == 1) {
        // Single K step, zero-padded K=1..3. All addresses valid for all lanes;
        // mask upper lane-half with selects (no predicated loads).
        const float xv = x[row];               // x is [n_rows, 1]
        const float wv = W[col];               // W is [1, FO]
        v2f a, b;
        a.x = (half == 0) ? xv : 0.0f;  a.y = 0.0f;
        b.x = (half == 0) ? wv : 0.0f;  b.y = 0.0f;
        c = __builtin_amdgcn_wmma_f32_16x16x4_f32(false, a, false, b,
                                                  (short)0, c, false, false);
    } else {
        static_assert(FI % 4 == 0, "FI must be a multiple of 4");
#pragma unroll
        for (int k0 = 0; k0 < FI; k0 += 4) {
            const int ka = k0 + half * 2;      // lane-half K offset (0 or 2)
            // A 16x4: lane L<16 -> {K=k0,k0+1}; L>=16 -> {K=k0+2,k0+3}; 8B-aligned pair.
            const v2f a = *reinterpret_cast<const v2f*>(x + row * FI + ka);
            // B 4x16: VGPR v, lane-half hf -> row K = k0 + v + 2*hf
            v2f b;
            b.x = W[ka * FO + col];
            b.y = W[(ka + 1) * FO + col];
            c = __builtin_amdgcn_wmma_f32_16x16x4_f32(false, a, false, b,
                                                      (short)0, c, false, false);
        }
    }

    // D 16x16 f32: VGPR v holds row M = v + 8*half, column = lane&15
    const int rbase = mt * 16 + half * 8;
#pragma unroll
    for (int v = 0; v < 8; ++v) {
        h[(rbase + v) * FO + col] = c[v];
    }
}

// ---------------- edge scatter: agg[dst] += h[src] * norm ----------------
// One thread per (edge, 4-feature chunk); consecutive threads cover consecutive
// 16B chunks of the same source row -> contiguous reads per edge row.
template <int FO>
__global__ void scatter_edges_kernel(const float* __restrict__ h,
                                     const float* __restrict__ norm,
                                     const int* __restrict__ src,
                                     const int* __restrict__ dst,
                                     float* __restrict__ agg,
                                     int n_edges) {
    constexpr int CH = FO / 4;                 // float4 chunks per row (power of two)
    const int idx = blockIdx.x * blockDim.x + threadIdx.x;
    if (idx >= n_edges * CH) return;
    const int e  = idx / CH;                   // compile-time shift
    const int ch = idx - e * CH;

    const float nrm = norm[e];
    const float4 hv = *reinterpret_cast<const float4*>(h + src[e] * FO + ch * 4);
    float* out = agg + dst[e] * FO + ch * 4;
    atomicAdd(out + 0, hv.x * nrm);
    atomicAdd(out + 1, hv.y * nrm);
    atomicAdd(out + 2, hv.z * nrm);
    atomicAdd(out + 3, hv.w * nrm);
}

// ---------------- bias + activation ----------------
// ACT: 0 = sigmoid, 1 = relu
template <int FO, int ACT>
__global__ void bias_act_kernel(const float* __restrict__ agg,
                                const float* __restrict__ bias,
                                float* __restrict__ out,
                                int total) {
    int i = blockIdx.x * blockDim.x + threadIdx.x;
    if (i >= total) return;
    float v = agg[i] + bias[i & (FO - 1)];     // FO is a power of two
    if constexpr (ACT == 0) v = 1.0f / (1.0f + expf(-v));
    else                    v = fmaxf(v, 0.0f);
    out[i] = v;
}

// ---------------- host launcher ----------------

static inline size_t align256(size_t x) { return (x + 255) & ~(size_t)255; }

template <int FI, int FO, int ACT>
static void run_layer(const float* x, const float* W, const float* b,
                      float* hbuf, float* aggbuf, float* outp,
                      int N, int E,
                      const float* nrm, const int* src, const int* dst,
                      hipStream_t stream) {
    const int T = 256;
    const int tiles = (N / 16) * (FO / 16);
    const int wpb   = T / 32;
    gemm_wmma_f32_kernel<FI, FO><<<(tiles + wpb - 1) / wpb, T, 0, stream>>>(x, W, hbuf, N);

    const int tot = N * FO;
    zero_f32_kernel<<<(tot + T - 1) / T, T, 0, stream>>>(aggbuf, tot);

    const int nthr = E * (FO / 4);
    scatter_edges_kernel<FO><<<(nthr + T - 1) / T, T, 0, stream>>>(hbuf, nrm, src, dst, aggbuf, E);

    bias_act_kernel<FO, ACT><<<(tot + T - 1) / T, T, 0, stream>>>(aggbuf, b, outp, tot);
}

extern "C" void kernel_launch(void* const* d_in, const int* in_sizes, int n_in,
                              void* d_out, int out_size, void* d_ws, size_t ws_size,
                              hipStream_t stream) {
    const float*      x_in = (const float*)d_in[0];
    const long long*  ei   = (const long long*)d_in[1];
    const float*      ew   = (const float*)d_in[2];
    const float* W1 = (const float*)d_in[3];  const float* b1 = (const float*)d_in[4];
    const float* W2 = (const float*)d_in[5];  const float* b2 = (const float*)d_in[6];
    const float* W3 = (const float*)d_in[7];  const float* b3 = (const float*)d_in[8];
    const float* W4 = (const float*)d_in[9];  const float* b4 = (const float*)d_in[10];
    const float* W5 = (const float*)d_in[11]; const float* b5 = (const float*)d_in[12];

    const int N = in_sizes[0];     // 100000 (x is [N,1])
    const int E = in_sizes[2];     // 1600000
    const int FMAX = 64;

    // workspace carve-up
    char* p = (char*)d_ws;
    float* deg  = (float*)p;                  p += align256((size_t)N * 4);
    float* nrm  = (float*)p;                  p += align256((size_t)E * 4);
    int*   src  = (int*)p;                    p += align256((size_t)E * 4);
    int*   dst  = (int*)p;                    p += align256((size_t)E * 4);
    float* bufA = (float*)p;                  p += align256((size_t)N * FMAX * 4);  // h
    float* bufB = (float*)p;                  p += align256((size_t)N * FMAX * 4);  // x / agg

    const int T = 256;
    // ---- norm precompute ----
    zero_f32_kernel<<<(N + T - 1) / T, T, 0, stream>>>(deg, N);
    prep_edges_kernel<<<(E + T - 1) / T, T, 0, stream>>>(ei, ew, src, dst, deg, E);
    dinv_kernel<<<(N + T - 1) / T, T, 0, stream>>>(deg, N);
    norm_kernel<<<(E + T - 1) / T, T, 0, stream>>>(deg, ew, src, dst, nrm, E);

    // ---- 5 GCN layers (sigmoid, relu, sigmoid, relu, sigmoid) ----
    run_layer<1,  16, 0>(x_in, W1, b1, bufA, bufB, bufB,          N, E, nrm, src, dst, stream);
    run_layer<16, 32, 1>(bufB, W2, b2, bufA, bufB, bufB,          N, E, nrm, src, dst, stream);
    run_layer<32, 64, 0>(bufB, W3, b3, bufA, bufB, bufB,          N, E, nrm, src, dst, stream);
    run_layer<64, 32, 1>(bufB, W4, b4, bufA, bufB, bufB,          N, E, nrm, src, dst, stream);
    run_layer<32, 16, 0>(bufB, W5, b5, bufA, bufB, (float*)d_out, N, E, nrm, src, dst, stream);
}